// FlexibleGNN_77884936946217
// MI455X (gfx1250) — compile-verified
//
#include <hip/hip_runtime.h>
#include <hip/hip_bf16.h>

#define GDIM 128            // feature dim D
#define EPS_RMS 1e-6f

typedef float v2f __attribute__((ext_vector_type(2)));
typedef float v8f __attribute__((ext_vector_type(8)));

// ---------------- degree / normalization ----------------
__global__ void deg_init_kernel(float* deg, int n) {
    int i = blockIdx.x * 256 + threadIdx.x;
    if (i < n) deg[i] = 1.0f;               // self-loop contribution
}

__global__ void deg_accum_kernel(const int* __restrict__ dst, float* deg, int ne) {
    int e = blockIdx.x * 256 + threadIdx.x;
    if (e < ne) atomicAdd(&deg[dst[e]], 1.0f);
}

__global__ void deg_finalize_kernel(float* deg, int n) {
    int i = blockIdx.x * 256 + threadIdx.x;
    if (i < n) deg[i] = rsqrtf(deg[i]);     // deg >= 1 always (self-loops)
}

// ---------------- fused (bias+relu)? + RMSNorm : wave per row ----------------
__global__ __launch_bounds__(256) void rms_kernel(const float* __restrict__ X,
                                                  const float* __restrict__ bias, // may be null
                                                  const float* __restrict__ g,
                                                  float* __restrict__ H, int nrows) {
    int row  = blockIdx.x * 8 + (threadIdx.x >> 5);
    if (row >= nrows) return;
    int lane = threadIdx.x & 31;

    float4 v = ((const float4*)(X + (size_t)row * GDIM))[lane];
    if (bias) {
        float4 b4 = ((const float4*)bias)[lane];
        v.x = fmaxf(v.x + b4.x, 0.0f);
        v.y = fmaxf(v.y + b4.y, 0.0f);
        v.z = fmaxf(v.z + b4.z, 0.0f);
        v.w = fmaxf(v.w + b4.w, 0.0f);
    }
    float ss = v.x * v.x + v.y * v.y + v.z * v.z + v.w * v.w;
    #pragma unroll
    for (int off = 16; off > 0; off >>= 1) ss += __shfl_xor(ss, off, 32);

    float scale = rsqrtf(ss * (1.0f / GDIM) + EPS_RMS);
    float4 g4 = ((const float4*)g)[lane];
    float4 o;
    o.x = v.x * scale * g4.x;
    o.y = v.y * scale * g4.y;
    o.z = v.z * scale * g4.z;
    o.w = v.w * scale * g4.w;
    ((float4*)(H + (size_t)row * GDIM))[lane] = o;
}

// ---------------- WMMA fp32 GEMM: T[nrows,128] = H[nrows,128] @ W[128,128] ----------------
// block = 256 threads = 8 waves; each wave owns a 16-row strip x 128 cols (8 tiles).
// W is staged in LDS pre-swizzled into B-fragment order:
//   Wsw[kg*512 + col*4 + ksub] = W[kg*4 + ksub][col]
// so a lane's B fragment (rows k0+koff, k0+koff+1 at column col) is a contiguous
// 8-byte pair -> single conflict-free ds_load_b64 into an even VGPR pair.
__global__ __launch_bounds__(256) void gemm_wmma_kernel(const float* __restrict__ H,
                                                        const float* __restrict__ W,
                                                        float* __restrict__ T, int nrows) {
    __shared__ float Wsw[(GDIM / 4) * GDIM * 4];   // 64 KB exactly

    int tid = threadIdx.x;
    // cooperative swizzled load of W into LDS
    for (int i = tid; i < (GDIM * GDIM) / 4; i += 256) {
        float4 w4 = ((const float4*)W)[i];
        int k = (i * 4) / GDIM;        // W row (K index)
        int c = (i * 4) % GDIM;        // W col base (N index)
        int kg = k >> 2, ks = k & 3;
        float* p = &Wsw[kg * 512 + ks];
        p[(c + 0) * 4] = w4.x;
        p[(c + 1) * 4] = w4.y;
        p[(c + 2) * 4] = w4.z;
        p[(c + 3) * 4] = w4.w;
    }
    __syncthreads();

    int wave    = tid >> 5;
    int lane    = tid & 31;
    int l16     = lane & 15;
    int halfsel = lane >> 4;              // 0: lanes 0-15, 1: lanes 16-31
    int koff    = halfsel * 2;            // A/B fragment K sub-offset per ISA layout

    int rowBase = blockIdx.x * 128 + wave * 16;
    int arow    = rowBase + l16;
    int arow_c  = arow < nrows ? arow : (nrows - 1);   // clamp loads; mask stores
    const float* hrow  = H + (size_t)arow_c * GDIM;
    const float* wbase = Wsw + l16 * 4 + koff;         // + kg*512 + nt*64

    v8f acc[8] = {};   // zero-initialized accumulators (8 column tiles)

    #pragma unroll 4
    for (int k0 = 0; k0 < GDIM; k0 += 4) {
        // A fragment (16x4 f32): lane half selects K pair -> one b64 global load
        v2f a = *(const v2f*)(hrow + k0 + koff);
        const float* wk = wbase + (k0 >> 2) * 512;
        #pragma unroll
        for (int nt = 0; nt < 8; ++nt) {
            // B fragment: contiguous pair -> single ds_load_b64
            v2f b = *(const v2f*)(wk + nt * 64);
            acc[nt] = __builtin_amdgcn_wmma_f32_16x16x4_f32(
                false, a, false, b, (short)0, acc[nt], false, false);
        }
    }

    // store C tiles: VGPR r -> row (rowBase + r + halfsel*8), col = nt*16 + l16
    for (int nt = 0; nt < 8; ++nt) {
        int col = nt * 16 + l16;
        #pragma unroll
        for (int r = 0; r < 8; ++r) {
            int row = rowBase + r + halfsel * 8;
            if (row < nrows) T[(size_t)row * GDIM + col] = acc[nt][r];
        }
    }
}

// ---------------- self-loop init: AGG[i,:] = T[i,:] * dinv[i]^2 ----------------
__global__ void agg_init_kernel(const float* __restrict__ T, const float* __restrict__ dinv,
                                float* __restrict__ AGG, int total) {
    int i = blockIdx.x * 256 + threadIdx.x;
    if (i < total) {
        int row = i >> 7;               // / 128
        float di = dinv[row];
        AGG[i] = T[i] * di * di;
    }
}

// ---------------- edge scatter: wave per edge, lane handles 4 features ----------------
__global__ __launch_bounds__(256) void scatter_kernel(const float* __restrict__ T,
                                                      const int* __restrict__ src,
                                                      const int* __restrict__ dst,
                                                      const float* __restrict__ dinv,
                                                      float* __restrict__ AGG, int ne) {
    int e = blockIdx.x * 8 + (threadIdx.x >> 5);
    if (e >= ne) return;
    int lane = threadIdx.x & 31;
    int s = src[e], d = dst[e];
    float c = dinv[s] * dinv[d];
    float4 t4 = ((const float4*)(T + (size_t)s * GDIM))[lane];
    float* out = AGG + (size_t)d * GDIM + lane * 4;
    atomicAdd(out + 0, t4.x * c);
    atomicAdd(out + 1, t4.y * c);
    atomicAdd(out + 2, t4.z * c);
    atomicAdd(out + 3, t4.w * c);
}

// ---------------- final bias + relu ----------------
__global__ void bias_relu_kernel(const float* __restrict__ A, const float* __restrict__ b,
                                 float* __restrict__ O, int total) {
    int i = blockIdx.x * 256 + threadIdx.x;
    if (i < total) {
        int dcol = i & (GDIM - 1);
        O[i] = fmaxf(A[i] + b[dcol], 0.0f);
    }
}

extern "C" void kernel_launch(void* const* d_in, const int* in_sizes, int n_in,
                              void* d_out, int out_size, void* d_ws, size_t ws_size,
                              hipStream_t stream) {
    const float* x  = (const float*)d_in[0];
    const int*   ei = (const int*)d_in[1];   // [2, E]: row0 = src, row1 = dst
    const float* Wm[3]   = {(const float*)d_in[2], (const float*)d_in[5], (const float*)d_in[8]};
    const float* bias[3] = {(const float*)d_in[3], (const float*)d_in[6], (const float*)d_in[9]};
    const float* gain[3] = {(const float*)d_in[4], (const float*)d_in[7], (const float*)d_in[10]};

    const int N = in_sizes[0] / GDIM;
    const int E = in_sizes[1] / 2;
    const int* src = ei;
    const int* dst = ei + E;

    // workspace layout: dinv[N] | B0[N*128] | B1[N*128]
    float* dinv = (float*)d_ws;
    size_t off  = ((size_t)N + 63) & ~(size_t)63;
    float* B0 = dinv + off;
    float* B1 = B0 + (size_t)N * GDIM;

    const int total = N * GDIM;

    // degrees -> dinv
    deg_init_kernel<<<(N + 255) / 256, 256, 0, stream>>>(dinv, N);
    deg_accum_kernel<<<(E + 255) / 256, 256, 0, stream>>>(dst, dinv, E);
    deg_finalize_kernel<<<(N + 255) / 256, 256, 0, stream>>>(dinv, N);

    const float* in_x = x;
    float* bufs[2] = {B0, B1};
    for (int l = 0; l < 3; ++l) {
        float* H = bufs[l & 1];          // rmsnorm output; later reused as AGG
        float* T = bufs[(l + 1) & 1];    // gemm output (messages)

        rms_kernel<<<(N + 7) / 8, 256, 0, stream>>>(
            in_x, (l == 0) ? nullptr : bias[l - 1], gain[l], H, N);
        gemm_wmma_kernel<<<(N + 127) / 128, 256, 0, stream>>>(H, Wm[l], T, N);
        agg_init_kernel<<<(total + 255) / 256, 256, 0, stream>>>(T, dinv, H, total);
        scatter_kernel<<<(E + 7) / 8, 256, 0, stream>>>(T, src, dst, dinv, H, E);

        in_x = H;                        // AGG becomes next layer's input
    }
    bias_relu_kernel<<<(total + 255) / 256, 256, 0, stream>>>(in_x, bias[2], (float*)d_out, total);
}